// KarplusStrong_30545807409680
// MI455X (gfx1250) — compile-verified
//
#include <hip/hip_runtime.h>
#include <stdint.h>

// ---------------------------------------------------------------------------
// Karplus-Strong on MI455X (gfx1250):
//   * one resident workgroup (4 wave32s) walks the sequential 32768-frame
//     filter recurrence; carry kept in LDS
//   * per-frame 512-pt FFT / iFFT done as Cooley-Tukey (16x32) with
//     V_WMMA_F32_16X16X4_F32 tiles (full fp32 precision)
//   * next-frame noise/tf rows prefetched to LDS with the Tensor Data Mover
//     (tensor_load_to_lds + s_wait_tensorcnt), double-buffered
//   * overlap-add fused: each frame writes its 256 finalized output samples
// ---------------------------------------------------------------------------

typedef float v2f __attribute__((ext_vector_type(2)));
typedef float v8f __attribute__((ext_vector_type(8)));
typedef unsigned int v4u __attribute__((ext_vector_type(4)));
typedef int v4i __attribute__((ext_vector_type(4)));
typedef int v8i __attribute__((ext_vector_type(8)));

#define KS_BLOCK 512
#define KS_HOP 256
#define KS_NT 128 // 4 waves of 32

// ---- WMMA fragment helpers (layouts per CDNA5 ISA 7.12.2) ------------------
// A 16x4 f32: v0 = {M=lane&15, K=k0 + 2*(lane>>4)}, v1 = same rows, K+1
__device__ __forceinline__ v2f ldA(const float* M, int ld, int k0, int lane) {
  int m = lane & 15;
  int kk = k0 + ((lane >> 4) << 1);
  v2f a;
  a.x = M[m * ld + kk];
  a.y = M[m * ld + kk + 1];
  return a;
}
// B 4x16 f32: assumed mirror of A (rows K0/K2 in v0, K1/K3 in v1; N across lanes)
__device__ __forceinline__ v2f ldB(const float* M, int ld, int k0, int n0, int lane) {
  int n = n0 + (lane & 15);
  int kk = k0 + ((lane >> 4) << 1);
  v2f b;
  b.x = M[kk * ld + n];
  b.y = M[(kk + 1) * ld + n];
  return b;
}
// C/D 16x16 f32: VGPR j holds M=j (lanes 0-15) and M=j+8 (lanes 16-31)
__device__ __forceinline__ void stD(float* M, int ld, int n0, int lane, v8f d) {
  int n = n0 + (lane & 15);
  int mb = (lane >> 4) << 3;
#pragma unroll
  for (int j = 0; j < 8; ++j) M[(mb + j) * ld + n] = d[j];
}

template <int K>
__device__ __forceinline__ v8f mmaTile(const float* A, int lda, const float* B, int ldb,
                                       int n0, int lane, v8f acc) {
#pragma unroll
  for (int k0 = 0; k0 < K; k0 += 4) {
    v2f a = ldA(A, lda, k0, lane);
    v2f b = ldB(B, ldb, k0, n0, lane);
    // 8 args: (neg_a, A, neg_b, B, c_mod, C, reuse_a, reuse_b)
    acc = __builtin_amdgcn_wmma_f32_16x16x4_f32(false, a, false, b, (short)0, acc,
                                                false, false);
  }
  return acc;
}

// ---- TDM: async 1-row tensor load (global f32 row -> LDS) ------------------
__device__ __forceinline__ void tdm_load_row(const float* gptr, float* lptr,
                                             unsigned int nelem, int lane) {
#if __has_builtin(__builtin_amdgcn_tensor_load_to_lds)
  unsigned long long ga = (unsigned long long)(uintptr_t)gptr;
  unsigned int la = (unsigned int)(uintptr_t)lptr; // flat->LDS: low 32 bits
  v4u g0;
  g0[0] = 1u;                                          // count=1, user descriptor
  g0[1] = la;                                          // lds_addr (bytes)
  g0[2] = (unsigned int)(ga & 0xFFFFFFFFu);            // global_addr[31:0]
  g0[3] = (unsigned int)((ga >> 32) & 0x1FFFFFFu)      // global_addr[56:32]
          | (2u << 30);                                // type=2 ("image")
  v8i g1;
  g1[0] = (int)(2u << 16);                             // data_size = 4B
  g1[1] = (int)((nelem & 0xFFFFu) << 16);              // tensor_dim0[15:0]
  g1[2] = (int)(((nelem >> 16) & 0xFFFFu) | (1u << 16)); // dim0 hi | tensor_dim1=1
  g1[3] = (int)((nelem & 0xFFFFu) << 16);              // tile_dim0 = nelem
  g1[4] = 1;                                           // tile_dim1 = 1
  g1[5] = (int)nelem;                                  // tensor_dim0_stride lo32
  g1[6] = 0;
  g1[7] = 0;
  v4i z4 = {0, 0, 0, 0};
#if __clang_major__ >= 23
  v8i z8 = {0, 0, 0, 0, 0, 0, 0, 0};
  __builtin_amdgcn_tensor_load_to_lds(g0, g1, z4, z4, z8, 0);
#else
  __builtin_amdgcn_tensor_load_to_lds(g0, g1, z4, z4, 0);
#endif
#else
  // fallback: cooperative copy by the issuing wave
  for (unsigned int i = (unsigned int)lane; i < nelem; i += 32u) lptr[i] = gptr[i];
#endif
}

__device__ __forceinline__ void tdm_wait_all() {
#if __has_builtin(__builtin_amdgcn_s_wait_tensorcnt)
  __builtin_amdgcn_s_wait_tensorcnt(0);
#endif
}

// ---------------------------------------------------------------------------
__global__ __launch_bounds__(KS_NT) void ks_kernel(const float* __restrict__ noise,
                                                   const float* __restrict__ env,
                                                   const float* __restrict__ tfr,
                                                   const float* __restrict__ tfi,
                                                   float* __restrict__ out,
                                                   int n_iter) {
  // factor matrices / state (about 58 KB static LDS)
  __shared__ float sCos32[1024], sSin32[1024];         // DFT-32 [n2][k2]
  __shared__ float sCos16[256], sSin16[256], sNSin16[256]; // DFT-16 [k1][n1]
  __shared__ float sTwc[512], sTws[512];               // twiddle [n1][k2], N=512
  __shared__ float sWnd[512];
  __shared__ float sCarry[512];
  __shared__ float sPrevHi[256];
  __shared__ float sNz[2][512];                        // noise double buffer
  __shared__ float sTr[2][260], sTi[2][260];           // tf double buffers
  __shared__ float sX[512];                            // [n1][k? n2] 16x32
  __shared__ float sYr[512], sYi[512], sZr[512], sZi[512];
  __shared__ float sORe[512], sOIm[512], sGr[512], sGi[512];
  __shared__ float sHr[512], sHi[512], sPr[512], sPi[512];
  __shared__ float sRA[512], sRB[512];

  const int tid = threadIdx.x;
  const int wid = tid >> 5;
  const int lane = tid & 31;

  // fire off the TDM prefetch of frame-0 inputs immediately (wave 0 only)
  if (wid == 0) {
    tdm_load_row(noise, sNz[0], 512u, lane);
    tdm_load_row(tfr, sTr[0], 257u, lane);
    tdm_load_row(tfi, sTi[0], 257u, lane);
  }

  // build trig / window tables while the DMA flies
  const float TWOPI = 6.28318530717958647692f;
  for (int i = tid; i < 1024; i += KS_NT) {
    int a = i >> 5, b = i & 31;
    float th = TWOPI * (float)((a * b) & 31) * (1.0f / 32.0f);
    sCos32[i] = cosf(th);
    sSin32[i] = sinf(th);
  }
  for (int i = tid; i < 256; i += KS_NT) {
    int a = i >> 4, b = i & 15;
    float th = TWOPI * (float)((a * b) & 15) * (1.0f / 16.0f);
    float c = cosf(th), s = sinf(th);
    sCos16[i] = c;
    sSin16[i] = s;
    sNSin16[i] = -s;
  }
  for (int i = tid; i < 512; i += KS_NT) {
    int n1 = i >> 5, k2 = i & 31;
    float th = TWOPI * (float)(n1 * k2) * (1.0f / 512.0f);
    sTwc[i] = cosf(th);
    sTws[i] = sinf(th);
    sWnd[i] = 0.54f - 0.46f * cosf(TWOPI * (float)i * (1.0f / 512.0f));
  }

  for (int f = 0; f < n_iter; ++f) {
    const int cb = f & 1;
    if (wid == 0) tdm_wait_all(); // frame f inputs landed in LDS
    __syncthreads();

    // prefetch next frame off the critical path
    if (wid == 0 && (f + 1) < n_iter) {
      tdm_load_row(noise + (size_t)(f + 1) * 512, sNz[cb ^ 1], 512u, lane);
      tdm_load_row(tfr + (size_t)(f + 1) * 257, sTr[cb ^ 1], 257u, lane);
      tdm_load_row(tfi + (size_t)(f + 1) * 257, sTi[cb ^ 1], 257u, lane);
    }

    // impulse + continuous excitation; fused overlap-add output for segment f
    float e = env[f];
    float e2 = e * e;
#pragma unroll
    for (int r = 0; r < 4; ++r) {
      int n = tid + r * KS_NT;
      float imp = (2.0f * sNz[cb][n] - 1.0f) * e2;
      float u = (f == 0 ? 0.0f : sCarry[n]) + imp;
      sX[(n & 15) * 32 + (n >> 4)] = u; // reshape n = n1 + 16*n2 -> [n1][n2]
      if (n < KS_HOP) {
        out[(size_t)f * KS_HOP + n] = (f == 0 ? 0.0f : sPrevHi[n]) + u;
      } else {
        sPrevHi[n - KS_HOP] = u; // same-thread ordering vs the reads above
      }
    }
    if (f == n_iter - 1) break; // `last` block is fully truncated by the OLA
    __syncthreads();

    // ---- forward stage 1: Y = X * DFT32  (Yr = X*C, Yi_st = X*S) ----------
    {
      const float* B = (wid < 2) ? sCos32 : sSin32;
      float* D = (wid < 2) ? sYr : sYi;
      int n0 = (wid & 1) << 4;
      v8f acc = {0, 0, 0, 0, 0, 0, 0, 0};
      acc = mmaTile<32>(sX, 32, B, 32, n0, lane, acc);
      stD(D, 32, n0, lane, acc);
    }
    __syncthreads();

    // ---- forward twiddle: (Yr - i*Yi_st) * (c - i*s) ----------------------
#pragma unroll
    for (int r = 0; r < 4; ++r) {
      int i = tid + r * KS_NT;
      float yr = sYr[i], yi = sYi[i];
      float c = sTwc[i], s = sTws[i];
      sZr[i] = yr * c - yi * s;
      sZi[i] = -(yr * s + yi * c);
    }
    __syncthreads();

    // ---- forward stage 3: Out = (cos16 - i*sin16) * Z ---------------------
    {
      bool re = (wid < 2);
      int n0 = (wid & 1) << 4;
      const float* B1 = re ? sZr : sZi;
      const float* A2 = re ? sSin16 : sNSin16;
      const float* B2 = re ? sZi : sZr;
      v8f acc = {0, 0, 0, 0, 0, 0, 0, 0};
      acc = mmaTile<16>(sCos16, 16, B1, 32, n0, lane, acc);
      acc = mmaTile<16>(A2, 16, B2, 32, n0, lane, acc);
      stD(re ? sORe : sOIm, 32, n0, lane, acc);
    }
    __syncthreads();

    // ---- pointwise transfer function (Hermitian-extended) -----------------
#pragma unroll
    for (int r = 0; r < 4; ++r) {
      int k = tid + r * KS_NT;
      float trv, tiv;
      if (k <= 256) {
        trv = sTr[cb][k];
        tiv = sTi[cb][k];
      } else {
        trv = sTr[cb][512 - k];
        tiv = -sTi[cb][512 - k];
      }
      float xr = sORe[k], xi = sOIm[k];
      sGr[k] = xr * trv - xi * tiv;
      sGi[k] = xr * tiv + xi * trv;
    }
    __syncthreads();

    // ---- inverse stage 1: H = (cos16 + i*sin16) * G -----------------------
    {
      bool re = (wid < 2);
      int n0 = (wid & 1) << 4;
      const float* B1 = re ? sGr : sGi;
      const float* A2 = re ? sNSin16 : sSin16;
      const float* B2 = re ? sGi : sGr;
      v8f acc = {0, 0, 0, 0, 0, 0, 0, 0};
      acc = mmaTile<16>(sCos16, 16, B1, 32, n0, lane, acc);
      acc = mmaTile<16>(A2, 16, B2, 32, n0, lane, acc);
      stD(re ? sHr : sHi, 32, n0, lane, acc);
    }
    __syncthreads();

    // ---- inverse twiddle: H * (c + i*s) -----------------------------------
#pragma unroll
    for (int r = 0; r < 4; ++r) {
      int i = tid + r * KS_NT;
      float hr = sHr[i], hi = sHi[i];
      float c = sTwc[i], s = sTws[i];
      sPr[i] = hr * c - hi * s;
      sPi[i] = hr * s + hi * c;
    }
    __syncthreads();

    // ---- inverse stage 3 (real part): Res = Pr*C32 - Pi*S32 ---------------
    {
      const float* A = (wid < 2) ? sPr : sPi;
      const float* B = (wid < 2) ? sCos32 : sSin32;
      float* D = (wid < 2) ? sRA : sRB;
      int n0 = (wid & 1) << 4;
      v8f acc = {0, 0, 0, 0, 0, 0, 0, 0};
      acc = mmaTile<32>(A, 32, B, 32, n0, lane, acc);
      stD(D, 32, n0, lane, acc);
    }
    __syncthreads();

    // ---- window + 1/N -> new carry (time-indexed) -------------------------
#pragma unroll
    for (int r = 0; r < 4; ++r) {
      int n = tid + r * KS_NT;
      int i = (n & 15) * 32 + (n >> 4);
      sCarry[n] = (sRA[i] - sRB[i]) * sWnd[n] * (1.0f / 512.0f);
    }
    // next-iteration top barrier orders carry writes with their readers
  }
}

// ---------------------------------------------------------------------------
extern "C" void kernel_launch(void* const* d_in, const int* in_sizes, int n_in,
                              void* d_out, int out_size, void* d_ws, size_t ws_size,
                              hipStream_t stream) {
  (void)n_in;
  (void)out_size;
  (void)d_ws;
  (void)ws_size;
  // setup_inputs order: x, noise, excitation_env, tf_real, tf_imag, piano_samples
  const float* noise = (const float*)d_in[1];
  const float* env = (const float*)d_in[2];
  const float* tfr = (const float*)d_in[3];
  const float* tfi = (const float*)d_in[4];
  const int n_iter = in_sizes[1] / KS_BLOCK; // noise is [n_iter, 512]
  ks_kernel<<<1, KS_NT, 0, stream>>>(noise, env, tfr, tfi, (float*)d_out, n_iter);
}